// EdgeAttentionEncoder_21809843929815
// MI455X (gfx1250) — compile-verified
//
#include <hip/hip_runtime.h>
#include <hip/hip_bf16.h>
#include <math.h>

typedef __attribute__((ext_vector_type(2))) float v2f;
typedef __attribute__((ext_vector_type(8))) float v8f;

// Problem dims (fixed by reference)
#define Bc   2
#define Nc   96
#define Dc   64
#define Hc   4
#define DKc  16
#define NNc  (Nc * Nc)           // 9216
#define ROWSc (Bc * NNc)         // 18432
#define PROJ_ELEMS  ((size_t)ROWSc * Dc)          // 1,179,648 floats
#define SCORE_ELEMS ((size_t)Bc * Hc * Nc * NNc)  // 7,077,888 floats

// ---------------------------------------------------------------------------
// Kernel 1: four fused input projections, out = state @ W^T + b
// One wave computes a 16x16 tile of ALL FOUR projections (A fragment reused).
// grid*block sized so every wave is full (EXEC all-ones, required by WMMA).
// ---------------------------------------------------------------------------
__global__ __launch_bounds__(256) void proj4_wmma(
    const float* __restrict__ state,
    const float* __restrict__ wlk, const float* __restrict__ blk,
    const float* __restrict__ wrk, const float* __restrict__ brk,
    const float* __restrict__ wlv, const float* __restrict__ blv,
    const float* __restrict__ wrv, const float* __restrict__ brv,
    float* __restrict__ lk, float* __restrict__ rk,
    float* __restrict__ lv, float* __restrict__ rv)
{
    const int lane = threadIdx.x & 31;
    const int wid  = blockIdx.x * 8 + (threadIdx.x >> 5); // 4608 waves total
    const int ct   = wid & 3;          // col tile (0..3) of 64 outputs
    const int rt   = wid >> 2;         // row tile (0..1151)
    const int r0   = rt << 4;
    const int c0   = ct << 4;

    const int l15  = lane & 15;        // A: row M ; B: col N ; D: col N
    const int koff = (lane >> 4) << 1; // K sub-offset 0 or 2

    const float* W[4]    = {wlk, wrk, wlv, wrv};
    const float* BIAS[4] = {blk, brk, blv, brv};
    float*       OUT[4]  = {lk, rk, lv, rv};

    v8f acc[4] = {{}, {}, {}, {}};

    const float* aBase = state + (size_t)(r0 + l15) * Dc + koff;
    #pragma unroll
    for (int k0 = 0; k0 < Dc; k0 += 4) {
        const v2f afrag = *(const v2f*)(aBase + k0);
        #pragma unroll
        for (int w = 0; w < 4; ++w) {
            const v2f bfrag = *(const v2f*)(W[w] + (size_t)(c0 + l15) * Dc + k0 + koff);
            acc[w] = __builtin_amdgcn_wmma_f32_16x16x4_f32(
                false, afrag, false, bfrag, (short)0, acc[w], false, false);
        }
    }

    const int srow = (lane >> 4) << 3; // 0 or 8: VGPR i holds rows i / i+8
    #pragma unroll
    for (int w = 0; w < 4; ++w) {
        const float bias = BIAS[w][c0 + l15];
        float* op = OUT[w] + (size_t)(r0 + srow) * Dc + c0 + l15;
        #pragma unroll
        for (int i = 0; i < 8; ++i)
            op[(size_t)i * Dc] = acc[w][i] + bias;
    }
}

// ---------------------------------------------------------------------------
// Kernel 2: scores[bh][a][x][y] = (1/sqrt(DK)) * sum_d lk[b,x,a,h,d]*rk[b,a,y,h,d]
// For fixed (b,h,a) this is a 96x96x16 GEMM -> one wave per 16x16 tile.
// ---------------------------------------------------------------------------
__global__ __launch_bounds__(256) void scores_wmma(
    const float* __restrict__ lk, const float* __restrict__ rk,
    float* __restrict__ sc)
{
    const int lane = threadIdx.x & 31;
    int wid = blockIdx.x * 8 + (threadIdx.x >> 5);  // 27648 waves total
    const int yt = wid % 6;  wid /= 6;
    const int xt = wid % 6;  wid /= 6;
    const int a  = wid % Nc; wid /= Nc;
    const int h  = wid % Hc;
    const int b  = wid / Hc;
    const int x0 = xt << 4, y0 = yt << 4;

    const int l15  = lane & 15;
    const int koff = (lane >> 4) << 1;

    // A fragment: rows = x, K = d;  B fragment: cols = y, K = d
    const float* aBase = lk + ((size_t)((b * Nc + (x0 + l15)) * Nc + a)) * Dc + h * DKc + koff;
    const float* bBase = rk + ((size_t)((b * Nc + a) * Nc + (y0 + l15))) * Dc + h * DKc + koff;

    v8f acc = {};
    #pragma unroll
    for (int k0 = 0; k0 < DKc; k0 += 4) {
        const v2f af = *(const v2f*)(aBase + k0);
        const v2f bf = *(const v2f*)(bBase + k0);
        acc = __builtin_amdgcn_wmma_f32_16x16x4_f32(
            false, af, false, bf, (short)0, acc, false, false);
    }

    const int srow = (lane >> 4) << 3;
    const int bh   = b * Hc + h;
    float* op = sc + ((size_t)((bh * Nc + a) * Nc + (x0 + srow))) * Nc + y0 + l15;
    #pragma unroll
    for (int i = 0; i < 8; ++i)
        op[(size_t)i * Nc] = acc[i] * 0.25f;   // 1/sqrt(16)
}

// ---------------------------------------------------------------------------
// Kernel 3: fused softmax over 'a' + triple contraction.
// One block per (b,h,x); thread = y. The 96x96 score panel for this (b,h,x)
// lives in LDS (36 KB of the 320 KB WGP LDS). Each thread touches only its
// own LDS column -> no barriers needed, scores read from global exactly once.
// ---------------------------------------------------------------------------
__global__ __launch_bounds__(96) void softmax_contract(
    const float* __restrict__ sc, const float* __restrict__ lv,
    const float* __restrict__ rv, float* __restrict__ xb)
{
    __shared__ float s[Nc * Nc];          // 36864 bytes
    const int y  = threadIdx.x;           // 0..95
    const int x  = blockIdx.x % Nc;
    const int bh = blockIdx.x / Nc;
    const int b  = bh >> 2;
    const int h  = bh & 3;

    // scores for (bh, a, x, y) : a-stride = NNc floats, coalesced in y
    const float* sp = sc + (size_t)bh * Nc * NNc + (size_t)x * Nc + y;

    float m = -INFINITY;
    for (int a = 0; a < Nc; ++a) {
        const float v = sp[(size_t)a * NNc];
        s[a * Nc + y] = v;
        m = fmaxf(m, v);
    }
    float sum = 0.f;
    for (int a = 0; a < Nc; ++a) {
        const float e = __expf(s[a * Nc + y] - m);
        s[a * Nc + y] = e;
        sum += e;
    }
    const float inv = 1.0f / sum;

    float acc[DKc];
    #pragma unroll
    for (int d = 0; d < DKc; ++d) acc[d] = 0.f;

    const float* lvBase = lv + ((size_t)(b * Nc + x) * Nc) * Dc + h * DKc; // +a*Dc (uniform)
    const float* rvBase = rv + ((size_t)b * NNc + y) * Dc + h * DKc;       // +a*Nc*Dc

    for (int a = 0; a < Nc; ++a) {
        const float att = s[a * Nc + y] * inv;
        const float* lvp = lvBase + (size_t)a * Dc;
        const float* rvp = rvBase + (size_t)a * (Nc * Dc);
        #pragma unroll
        for (int d = 0; d < DKc; ++d)
            acc[d] = fmaf(att * lvp[d], rvp[d], acc[d]);
    }

    float* op = xb + ((size_t)(b * Nc + x) * Nc + y) * Dc + h * DKc;
    #pragma unroll
    for (int d = 0; d < DKc; ++d) op[d] = acc[d];
}

// ---------------------------------------------------------------------------
// Kernel 4: output projection, d_out = xb @ w_out^T + b_out (WMMA f32)
// ---------------------------------------------------------------------------
__global__ __launch_bounds__(256) void outproj_wmma(
    const float* __restrict__ xb, const float* __restrict__ wout,
    const float* __restrict__ bout, float* __restrict__ out)
{
    const int lane = threadIdx.x & 31;
    const int wid  = blockIdx.x * 8 + (threadIdx.x >> 5);
    const int ct   = wid & 3;
    const int rt   = wid >> 2;
    const int r0   = rt << 4;
    const int c0   = ct << 4;

    const int l15  = lane & 15;
    const int koff = (lane >> 4) << 1;

    v8f acc = {};
    const float* aBase = xb   + (size_t)(r0 + l15) * Dc + koff;
    const float* bBase = wout + (size_t)(c0 + l15) * Dc + koff;
    #pragma unroll
    for (int k0 = 0; k0 < Dc; k0 += 4) {
        const v2f af = *(const v2f*)(aBase + k0);
        const v2f bf = *(const v2f*)(bBase + k0);
        acc = __builtin_amdgcn_wmma_f32_16x16x4_f32(
            false, af, false, bf, (short)0, acc, false, false);
    }

    const int srow = (lane >> 4) << 3;
    const float bias = bout[c0 + l15];
    float* op = out + (size_t)(r0 + srow) * Dc + c0 + l15;
    #pragma unroll
    for (int i = 0; i < 8; ++i)
        op[(size_t)i * Dc] = acc[i] + bias;
}

// ---------------------------------------------------------------------------
extern "C" void kernel_launch(void* const* d_in, const int* in_sizes, int n_in,
                              void* d_out, int out_size, void* d_ws, size_t ws_size,
                              hipStream_t stream) {
    (void)in_sizes; (void)n_in; (void)out_size; (void)ws_size;

    const float* state = (const float*)d_in[0];
    const float* w_lk  = (const float*)d_in[1];
    const float* b_lk  = (const float*)d_in[2];
    const float* w_rk  = (const float*)d_in[3];
    const float* b_rk  = (const float*)d_in[4];
    const float* w_lv  = (const float*)d_in[5];
    const float* b_lv  = (const float*)d_in[6];
    const float* w_rv  = (const float*)d_in[7];
    const float* b_rv  = (const float*)d_in[8];
    const float* w_out = (const float*)d_in[9];
    const float* b_out = (const float*)d_in[10];
    float* out = (float*)d_out;

    float* ws = (float*)d_ws;
    float* lk = ws;
    float* rk = lk + PROJ_ELEMS;
    float* lv = rk + PROJ_ELEMS;
    float* rv = lv + PROJ_ELEMS;
    float* sc = rv + PROJ_ELEMS;     // SCORE_ELEMS floats
    float* xb = sc + SCORE_ELEMS;    // PROJ_ELEMS floats

    // 1) four input projections: 4608 waves = 576 blocks x 8 waves (exact)
    proj4_wmma<<<576, 256, 0, stream>>>(state,
        w_lk, b_lk, w_rk, b_rk, w_lv, b_lv, w_rv, b_rv,
        lk, rk, lv, rv);

    // 2) per-(b,h,a) 96x96x16 GEMMs: 27648 waves = 3456 blocks x 8 waves
    scores_wmma<<<3456, 256, 0, stream>>>(lk, rk, sc);

    // 3) fused softmax-over-a + triple contraction: block per (b,h,x)
    softmax_contract<<<Bc * Hc * Nc, Nc, 0, stream>>>(sc, lv, rv, xb);

    // 4) output projection: 4608 waves = 576 blocks x 8 waves
    outproj_wmma<<<576, 256, 0, stream>>>(xb, w_out, b_out, out);
}